// ProteinInteractionPredictor_13898514169957
// MI455X (gfx1250) — compile-verified
//
#include <hip/hip_runtime.h>
#include <math.h>

// ---------------------------------------------------------------------------
// GCN protein-interaction predictor for MI455X (gfx1250, wave32).
//
// Roofline: edge gather/scatter (~5 GB over 6 conv layers) bounds runtime at
// ~0.25 ms @ 23.3 TB/s. GEMMs (6 x 1.64 GFLOP) are negligible but are the
// matrix-shaped work, done with V_WMMA_F32_16X16X4_F32 (exact fp32).
// ---------------------------------------------------------------------------

typedef __attribute__((ext_vector_type(2))) float v2f;
typedef __attribute__((ext_vector_type(8))) float v8f;

#define FEAT 128

// ---------------------------------------------------------------------------
// Degree: deg[n] = in-degree (self loop added later as +1 in norm kernel)
// ---------------------------------------------------------------------------
__global__ void deg_kernel(const int* __restrict__ dst, int* __restrict__ deg, int E) {
    int t = blockIdx.x * blockDim.x + threadIdx.x;
    if (t < E) atomicAdd(&deg[dst[t]], 1);
}

__global__ void norm_kernel(const int* __restrict__ deg, float* __restrict__ inv, int N) {
    int t = blockIdx.x * blockDim.x + threadIdx.x;
    if (t < N) inv[t] = rsqrtf((float)(deg[t] + 1));   // +1 = self loop; always > 0
}

// ---------------------------------------------------------------------------
// H = X @ W   (X: nRows x 128, W: 128 x 128, fp32)
// One block = 16 rows of X staged in LDS; 8 waves, each wave owns a 16x16
// output tile; K=128 consumed as 32 chained V_WMMA_F32_16X16X4_F32 ops.
//
// Lane mapping (wave32, per ISA 7.12.2):
//   A 16x4 f32 : lane l holds row m=l&15, K pair = kBase + (l>>4)*2
//   B 4x16 f32 : lane l holds col n=l&15, same K pair
//   C/D 16x16  : lane l, VGPR j -> M = j + 8*(l>>4), N = l&15
// ---------------------------------------------------------------------------
__global__ __launch_bounds__(256)
void gemm128_wmma(const float* __restrict__ X, const float* __restrict__ W,
                  float* __restrict__ H, int nRows) {
    __shared__ float As[16 * FEAT];          // 8 KB A tile

    const int mBase = blockIdx.x * 16;
    const int t = threadIdx.x;

    // Cooperative coalesced load of the 16x128 A tile (2048 floats, float4).
    const float4* Xv = (const float4*)(X + (size_t)mBase * FEAT);
    float4* Av = (float4*)As;
    Av[t]       = Xv[t];
    Av[t + 256] = Xv[t + 256];
    __syncthreads();

    const int lane    = t & 31;
    const int wave    = t >> 5;
    const int colBase = wave * 16;           // 8 waves cover all 128 output cols
    const int mn      = lane & 15;           // row index for A, col index for B
    const int kHalf   = (lane >> 4) * 2;     // lanes 16-31 hold the upper K pair

    v8f c = {};
    #pragma unroll
    for (int k = 0; k < FEAT; k += 4) {
        const int kk = k + kHalf;
        v2f a, b;
        a.x = As[mn * FEAT + kk];
        a.y = As[mn * FEAT + kk + 1];
        b.x = W[(size_t)kk       * FEAT + colBase + mn];
        b.y = W[(size_t)(kk + 1) * FEAT + colBase + mn];
        c = __builtin_amdgcn_wmma_f32_16x16x4_f32(
                /*neg_a=*/false, a, /*neg_b=*/false, b,
                /*c_mod=*/(short)0, c, /*reuse_a=*/false, /*reuse_b=*/false);
    }

    const int rowBase = mBase + (lane >> 4) * 8;
    #pragma unroll
    for (int j = 0; j < 8; ++j)
        H[(size_t)(rowBase + j) * FEAT + colBase + mn] = c[j];
}

// ---------------------------------------------------------------------------
// Edge scatter: acc[dst] += h[src] * inv[src]*inv[dst]
// One wave per edge: 32 lanes x float4 = the full 128-float feature row.
// Gather of h[src] is fully coalesced; scatter uses hw global_atomic_add_f32.
// ---------------------------------------------------------------------------
__global__ __launch_bounds__(256)
void edge_scatter(const float* __restrict__ H, const int* __restrict__ src,
                  const int* __restrict__ dst, const float* __restrict__ inv,
                  float* __restrict__ acc, int E) {
    int t = blockIdx.x * blockDim.x + threadIdx.x;
    int e = t >> 5;
    if (e >= E) return;
    int q = (t & 31) << 2;                   // feature offset, float4 granule

    int s = src[e], d = dst[e];
    float w = inv[s] * inv[d];

    float4 v = *(const float4*)(H + (size_t)s * FEAT + q);
    float* o = acc + (size_t)d * FEAT + q;
    unsafeAtomicAdd(o + 0, v.x * w);
    unsafeAtomicAdd(o + 1, v.y * w);
    unsafeAtomicAdd(o + 2, v.z * w);
    unsafeAtomicAdd(o + 3, v.w * w);
}

// ---------------------------------------------------------------------------
// Finalize conv layer in place: acc = relu(acc + h*inv^2 (self loop) + bias)
// ---------------------------------------------------------------------------
__global__ void conv_finalize(float* __restrict__ acc, const float* __restrict__ H,
                              const float* __restrict__ inv, const float* __restrict__ bias,
                              int N) {
    int t = blockIdx.x * blockDim.x + threadIdx.x;
    if (t >= N * FEAT) return;
    int n = t >> 7, f = t & (FEAT - 1);
    float iv = inv[n];
    float v = acc[t] + H[t] * iv * iv + bias[f];
    acc[t] = fmaxf(v, 0.0f);
}

// ---------------------------------------------------------------------------
// Global mean pool (sum part): pool[f] += sum over node-stride of X[n][f]
// ---------------------------------------------------------------------------
__global__ void pool_kernel(const float* __restrict__ X, float* __restrict__ pool, int N) {
    int f = threadIdx.x;                     // blockDim = 128
    float s = 0.0f;
    for (int n = blockIdx.x; n < N; n += gridDim.x)
        s += X[(size_t)n * FEAT + f];
    unsafeAtomicAdd(&pool[f], s);
}

// ---------------------------------------------------------------------------
// Head: meta MLPs + concat + classifier, one 256-thread block (trivial work).
// ---------------------------------------------------------------------------
__global__ __launch_bounds__(256)
void head_kernel(const float* __restrict__ meta_a, const float* __restrict__ meta_b,
                 const float* __restrict__ fc1_w, const float* __restrict__ fc1_b,
                 const float* __restrict__ fc2_w, const float* __restrict__ fc2_b,
                 const float* __restrict__ poolA, const float* __restrict__ poolB,
                 float invN,
                 const float* __restrict__ fcc_w,  const float* __restrict__ fcc_b,
                 const float* __restrict__ fcc2_w, const float* __restrict__ fcc2_b,
                 const float* __restrict__ out_w,  const float* __restrict__ out_b,
                 float* __restrict__ out) {
    __shared__ float m[1024];
    __shared__ float t1[256];
    __shared__ float c[512];
    __shared__ float d1[256];
    __shared__ float d2[128];
    __shared__ float red[256];
    const int t = threadIdx.x;

    // meta branches -> c[0:128] (a), c[128:256] (b)
    for (int pass = 0; pass < 2; ++pass) {
        const float* meta = pass ? meta_b : meta_a;
        for (int i = t; i < 1024; i += 256) m[i] = meta[i];
        __syncthreads();
        float s = fc1_b[t];
        for (int k = 0; k < 1024; ++k) s += m[k] * fc1_w[k * 256 + t];
        t1[t] = fmaxf(s, 0.0f);
        __syncthreads();
        if (t < 128) {
            float s2 = fc2_b[t];
            for (int k = 0; k < 256; ++k) s2 += t1[k] * fc2_w[k * 128 + t];
            c[pass * 128 + t] = fmaxf(s2, 0.0f);
        }
        __syncthreads();
    }
    // graph branches -> c[256:384] (a), c[384:512] (b)
    if (t < 128) c[256 + t] = poolA[t] * invN;
    else         c[256 + t] = poolB[t - 128] * invN;
    __syncthreads();

    {   // fcc: 512 -> 256, relu
        float s = fcc_b[t];
        for (int k = 0; k < 512; ++k) s += c[k] * fcc_w[k * 256 + t];
        d1[t] = fmaxf(s, 0.0f);
    }
    __syncthreads();
    if (t < 128) {                            // fcc2: 256 -> 128, relu
        float s = fcc2_b[t];
        for (int k = 0; k < 256; ++k) s += d1[k] * fcc2_w[k * 128 + t];
        d2[t] = fmaxf(s, 0.0f);
    }
    __syncthreads();
    red[t] = (t < 128) ? d2[t] * out_w[t] : 0.0f;
    __syncthreads();
    for (int s = 128; s > 0; s >>= 1) {
        if (t < s) red[t] += red[t + s];
        __syncthreads();
    }
    if (t == 0) out[0] = 1.0f / (1.0f + expf(-(red[0] + out_b[0])));
}

// ---------------------------------------------------------------------------
// Host-side orchestration
// ---------------------------------------------------------------------------
extern "C" void kernel_launch(void* const* d_in, const int* in_sizes, int n_in,
                              void* d_out, int out_size, void* d_ws, size_t ws_size,
                              hipStream_t stream) {
    const float* meta_a = (const float*)d_in[0];
    const float* meta_b = (const float*)d_in[1];
    const float* x_a    = (const float*)d_in[2];
    const int*   ei_a   = (const int*)  d_in[3];
    const float* x_b    = (const float*)d_in[4];
    const int*   ei_b   = (const int*)  d_in[5];
    const float* fc1_w  = (const float*)d_in[6];
    const float* fc1_b  = (const float*)d_in[7];
    const float* fc2_w  = (const float*)d_in[8];
    const float* fc2_b  = (const float*)d_in[9];
    const float* gcn1_w = (const float*)d_in[10];
    const float* gcn1_b = (const float*)d_in[11];
    const float* gcn2_w = (const float*)d_in[12];
    const float* gcn2_b = (const float*)d_in[13];
    const float* fcc_w  = (const float*)d_in[14];
    const float* fcc_b  = (const float*)d_in[15];
    const float* fcc2_w = (const float*)d_in[16];
    const float* fcc2_b = (const float*)d_in[17];
    const float* out_w  = (const float*)d_in[18];
    const float* out_b  = (const float*)d_in[19];

    const int N = in_sizes[2] / FEAT;        // 50000
    const int E = in_sizes[3] / 2;           // 800000

    // Workspace layout (~52 MB)
    float* bufH  = (float*)d_ws;             // N*128 f32
    float* bufX  = bufH + (size_t)N * FEAT;  // N*128 f32
    float* inv   = bufX + (size_t)N * FEAT;  // N f32
    int*   deg   = (int*)(inv + N);          // N i32
    float* poolA = (float*)(deg + N);        // 128 f32
    float* poolB = poolA + FEAT;             // 128 f32

    const size_t nodeBytes = (size_t)N * FEAT * sizeof(float);
    const int gemmBlocks   = (N + 15) / 16;
    const int scatBlocks   = (E * 32 + 255) / 256;
    const int finBlocks    = (N * FEAT + 255) / 256;

    const float* graphX[2]  = { x_a, x_b };
    const int*   graphEI[2] = { ei_a, ei_b };
    float*       pools[2]   = { poolA, poolB };

    for (int g = 0; g < 2; ++g) {
        const int* src = graphEI[g];
        const int* dst = graphEI[g] + E;

        // degrees + symmetric normalization
        hipMemsetAsync(deg, 0, (size_t)N * sizeof(int), stream);
        deg_kernel <<<(E + 255) / 256, 256, 0, stream>>>(dst, deg, E);
        norm_kernel<<<(N + 255) / 256, 256, 0, stream>>>(deg, inv, N);

        const float* xin = graphX[g];
        for (int layer = 0; layer < 3; ++layer) {
            const float* W = (layer == 0) ? gcn1_w : gcn2_w;
            const float* B = (layer == 0) ? gcn1_b : gcn2_b;

            gemm128_wmma<<<gemmBlocks, 256, 0, stream>>>(xin, W, bufH, N);
            hipMemsetAsync(bufX, 0, nodeBytes, stream);   // stream-ordered: after gemm read
            edge_scatter<<<scatBlocks, 256, 0, stream>>>(bufH, src, dst, inv, bufX, E);
            conv_finalize<<<finBlocks, 256, 0, stream>>>(bufX, bufH, inv, B, N);
            xin = bufX;
        }

        hipMemsetAsync(pools[g], 0, FEAT * sizeof(float), stream);
        pool_kernel<<<256, FEAT, 0, stream>>>(bufX, pools[g], N);
    }

    head_kernel<<<1, 256, 0, stream>>>(meta_a, meta_b, fc1_w, fc1_b, fc2_w, fc2_b,
                                       poolA, poolB, 1.0f / (float)N,
                                       fcc_w, fcc_b, fcc2_w, fcc2_b,
                                       out_w, out_b, (float*)d_out);
}